// WeightedPooling_54236847013950
// MI455X (gfx1250) — compile-verified
//
#include <hip/hip_runtime.h>

typedef float v2f __attribute__((ext_vector_type(2)));
typedef float v8f __attribute__((ext_vector_type(8)));

#define MAT_N 64
#define LD 68              // padded LDS row stride (floats) -> conflict-free A-frag loads
#define CH 16
#define BATCH 512
#define LOG_DEG 26         // Chebyshev degree for log on [LOG_A, LOG_B]
#define CHEB_NODES 64
#define LOG_A 0.75f
#define LOG_B 7.0f
#define EXP_TAYLOR 9
#define EXP_SQUARINGS 4    // exp(M) = (exp(M/16))^16

// ---------------- WMMA fragment helpers (f32 16x16x4) ----------------
// A 16x4 f32: lane l holds row m = l&15, k-pair kb = 2*(l>>4): {A[m][kb], A[m][kb+1]}
__device__ __forceinline__ v2f load_a_frag(const float* M, int row0, int k0, int lane) {
  int m  = row0 + (lane & 15);
  int kb = k0 + ((lane >> 4) << 1);
  const float* p = M + m * LD + kb;
  v2f a; a.x = p[0]; a.y = p[1];
  return a;
}
// B 4x16 f32: lane l holds col n = l&15, rows kb, kb+1 with kb = k0 + 2*(l>>4)
__device__ __forceinline__ v2f load_b_frag(const float* M, int k0, int col0, int lane) {
  int n  = col0 + (lane & 15);
  int kb = k0 + ((lane >> 4) << 1);
  v2f b; b.x = M[kb * LD + n]; b.y = M[(kb + 1) * LD + n];
  return b;
}
// C/D 16x16 f32: vgpr r holds row (r + 8*(l>>4)), col = l&15
__device__ __forceinline__ v8f load_c_tile(const float* M, int row0, int col0, int lane) {
  int col = col0 + (lane & 15);
  int rb  = row0 + ((lane >> 4) << 3);
  v8f c;
#pragma unroll
  for (int r = 0; r < 8; ++r) c[r] = M[(rb + r) * LD + col];
  return c;
}
__device__ __forceinline__ v8f load_c_tile_scaled(const float* M, int row0, int col0, int lane,
                                                  float scale) {
  int col = col0 + (lane & 15);
  int rb  = row0 + ((lane >> 4) << 3);
  v8f c;
#pragma unroll
  for (int r = 0; r < 8; ++r) c[r] = M[(rb + r) * LD + col] * scale;
  return c;
}
__device__ __forceinline__ void store_c_tile(float* M, int row0, int col0, int lane, v8f c) {
  int col = col0 + (lane & 15);
  int rb  = row0 + ((lane >> 4) << 3);
#pragma unroll
  for (int r = 0; r < 8; ++r) M[(rb + r) * LD + col] = c[r];
}
__device__ __forceinline__ void add_diag(v8f& c, int tm, int tn, int lane, float val) {
  if (tm == tn) {
    if (lane < 8)        c[lane] += val;
    else if (lane >= 24) c[lane - 24] += val;
  }
}
__device__ __forceinline__ v8f wmma_k64(const float* A, const float* B,
                                        int tm, int tn, int lane, v8f acc) {
#pragma unroll
  for (int kk = 0; kk < MAT_N; kk += 4) {
    v2f a = load_a_frag(A, tm * 16, kk, lane);
    v2f b = load_b_frag(B, kk, tn * 16, lane);
    acc = __builtin_amdgcn_wmma_f32_16x16x4_f32(false, a, false, b, (short)0, acc, false, false);
  }
  return acc;
}

// ---------------- sparsemax of the 16x16 weight matrix ----------------
__global__ void sparsemax_kernel(const float* __restrict__ w, float* __restrict__ wsm) {
  int i = threadIdx.x;
  if (i >= CH) return;
  float z[CH], zs[CH];
#pragma unroll
  for (int j = 0; j < CH; ++j) { z[j] = w[i * CH + j]; zs[j] = z[j]; }
  // insertion sort, descending
  for (int a = 1; a < CH; ++a) {
    float key = zs[a]; int b = a - 1;
    while (b >= 0 && zs[b] < key) { zs[b + 1] = zs[b]; --b; }
    zs[b + 1] = key;
  }
  float cum = 0.f, tau = 0.f;
  for (int k = 0; k < CH; ++k) {
    cum += zs[k];
    if (1.0f + (float)(k + 1) * zs[k] > cum) tau = (cum - 1.0f) / (float)(k + 1);
  }
#pragma unroll
  for (int j = 0; j < CH; ++j) wsm[i * CH + j] = fmaxf(z[j] - tau, 0.f);
}

// ---------------- logm via matrix Chebyshev (spectrum in [LOG_A, LOG_B]) ----------------
// Sign-folded recurrence: U_k = (-1)^k T_k, so U_k = (-2Y)*U_{k-1} + U_{k-2}
// (no C-operand negation needed), and S = sum_k [(-1)^k c_k] * U_k.
__launch_bounds__(256)
__global__ void logm_kernel(const float* __restrict__ x, float* __restrict__ logs) {
  __shared__ float Y2n[MAT_N * LD];  // -2*Y, Y = affinely scaled input on [-1,1]
  __shared__ float Ta[MAT_N * LD];
  __shared__ float Tb[MAT_N * LD];
  __shared__ float coef[LOG_DEG + 1];  // (-1)^k * c_k

  const int tid = threadIdx.x;
  const int lane = tid & 31;
  const int wave = tid >> 5;
  const float* X   = x    + (size_t)blockIdx.x * (MAT_N * MAT_N);
  float*       OUT = logs + (size_t)blockIdx.x * (MAT_N * MAT_N);

  // Chebyshev coefficients of log on [LOG_A, LOG_B] via 64-node DCT (on device),
  // with the (-1)^k sign of the U-recurrence folded in.
  if (tid <= LOG_DEG) {
    float s = 0.f;
    for (int j = 0; j < CHEB_NODES; ++j) {
      float th = 3.14159265358979f * ((float)j + 0.5f) / (float)CHEB_NODES;
      float xj = 0.5f * ((LOG_B + LOG_A) + (LOG_B - LOG_A) * __cosf(th));
      s += __logf(xj) * __cosf((float)tid * th);
    }
    float c = s * (2.0f / (float)CHEB_NODES);
    if (tid == 0) c *= 0.5f;
    coef[tid] = (tid & 1) ? -c : c;
  }

  // Y2n = -(alpha*X + beta*I) ; U1 = -Y = 0.5*Y2n ; U0 = I
  {
    const float alpha = 4.0f / (LOG_B - LOG_A);
    const float beta  = -2.0f * (LOG_B + LOG_A) / (LOG_B - LOG_A);
    int r  = tid >> 2;
    int c0 = (tid & 3) * 16;
    const float4* src = (const float4*)(X + r * MAT_N + c0);
#pragma unroll
    for (int q = 0; q < 4; ++q) {
      float4 v = src[q];
      float vv[4] = {v.x, v.y, v.z, v.w};
#pragma unroll
      for (int e = 0; e < 4; ++e) {
        int c = c0 + q * 4 + e;
        float y2n = -(vv[e] * alpha + ((r == c) ? beta : 0.f));
        Y2n[r * LD + c] = y2n;
        Ta[r * LD + c]  = 0.5f * y2n;           // U1 = -Y
        Tb[r * LD + c]  = (r == c) ? 1.f : 0.f;  // U0 = I
      }
    }
  }
  __syncthreads();

  const int tm0 = wave >> 2;  // this wave's tile rows: tm0, tm0+2
  const int tn  = wave & 3;

  // Register-resident result: S = d0*I + d1*U1  (d_k = (-1)^k c_k)
  v8f S[2];
#pragma unroll
  for (int t = 0; t < 2; ++t) {
    int tm = tm0 + t * 2;
    S[t] = load_c_tile_scaled(Ta, tm * 16, tn * 16, lane, coef[1]);
    add_diag(S[t], tm, tn, lane, coef[0]);
  }

  float* Ucur = Ta;   // U_{k-1}
  float* Uprev = Tb;  // U_{k-2}
  for (int k = 2; k <= LOG_DEG; ++k) {
    v8f acc[2];
#pragma unroll
    for (int t = 0; t < 2; ++t) {
      int tm = tm0 + t * 2;
      acc[t] = load_c_tile(Uprev, tm * 16, tn * 16, lane);  // + U_{k-2}, no negation
      acc[t] = wmma_k64(Y2n, Ucur, tm, tn, lane, acc[t]);   // + (-2Y)*U_{k-1}
      S[t] += coef[k] * acc[t];
    }
    __syncthreads();                      // all reads of Uprev/Ucur done
#pragma unroll
    for (int t = 0; t < 2; ++t)
      store_c_tile(Uprev, tm0 * 16 + t * 32, tn * 16, lane, acc[t]); // U_k over U_{k-2}
    __syncthreads();
    float* tmp = Uprev; Uprev = Ucur; Ucur = tmp;
  }

  // write logm result
#pragma unroll
  for (int t = 0; t < 2; ++t) {
    int tm  = tm0 + t * 2;
    int row = tm * 16 + ((lane >> 4) << 3);
    int col = tn * 16 + (lane & 15);
#pragma unroll
    for (int r = 0; r < 8; ++r) OUT[(row + r) * MAT_N + col] = S[t][r];
  }
}

// ---------------- mix + expm (scaling & squaring, Taylor-9 Horner) ----------------
__launch_bounds__(256)
__global__ void expmix_kernel(const float* __restrict__ logs,
                              const float* __restrict__ wsm_g,
                              float* __restrict__ out) {
  __shared__ float Xs[MAT_N * LD];   // mixed/2^s
  __shared__ float Rs[MAT_N * LD];   // Horner accumulator / result
  __shared__ float wsm[CH * CH];

  const int tid = threadIdx.x;
  const int lane = tid & 31;
  const int wave = tid >> 5;
  const int b = blockIdx.x >> 4;
  const int i = blockIdx.x & 15;

  wsm[tid] = wsm_g[tid];             // 256 threads == CH*CH
  __syncthreads();

  // X = (1/16) * sum_c wsm[i][c] * logs[b][c]
  {
    int r  = tid >> 2;
    int c0 = (tid & 3) * 16;
    float acc[16];
#pragma unroll
    for (int e = 0; e < 16; ++e) acc[e] = 0.f;
    for (int c = 0; c < CH; ++c) {
      float wv = wsm[i * CH + c];
      const float4* src =
          (const float4*)(logs + (size_t)(b * CH + c) * (MAT_N * MAT_N) + r * MAT_N + c0);
#pragma unroll
      for (int q = 0; q < 4; ++q) {
        float4 v = src[q];
        acc[q * 4 + 0] += wv * v.x;
        acc[q * 4 + 1] += wv * v.y;
        acc[q * 4 + 2] += wv * v.z;
        acc[q * 4 + 3] += wv * v.w;
      }
    }
    const float sc = 1.0f / (float)(1 << EXP_SQUARINGS);
#pragma unroll
    for (int e = 0; e < 16; ++e) Xs[r * LD + c0 + e] = acc[e] * sc;
    // Horner seed: R = I + X/EXP_TAYLOR
#pragma unroll
    for (int e = 0; e < 16; ++e) {
      int c = c0 + e;
      Rs[r * LD + c] = acc[e] * (sc / (float)EXP_TAYLOR) + ((r == c) ? 1.f : 0.f);
    }
  }
  __syncthreads();

  const int tm0 = wave >> 2;
  const int tn  = wave & 3;

  // Horner: R <- I + (X*R)/k, k = 8..1
  for (int k = EXP_TAYLOR - 1; k >= 1; --k) {
    float inv = 1.0f / (float)k;
    v8f acc[2];
#pragma unroll
    for (int t = 0; t < 2; ++t) {
      int tm = tm0 + t * 2;
      v8f z = {};
      z = wmma_k64(Xs, Rs, tm, tn, lane, z);
      acc[t] = z * inv;
      add_diag(acc[t], tm, tn, lane, 1.0f);
    }
    __syncthreads();
#pragma unroll
    for (int t = 0; t < 2; ++t)
      store_c_tile(Rs, tm0 * 16 + t * 32, tn * 16, lane, acc[t]);
    __syncthreads();
  }

  // squarings: R <- R*R, 4 times
  for (int s2 = 0; s2 < EXP_SQUARINGS; ++s2) {
    v8f acc[2];
#pragma unroll
    for (int t = 0; t < 2; ++t) {
      int tm = tm0 + t * 2;
      v8f z = {};
      acc[t] = wmma_k64(Rs, Rs, tm, tn, lane, z);
    }
    __syncthreads();
#pragma unroll
    for (int t = 0; t < 2; ++t)
      store_c_tile(Rs, tm0 * 16 + t * 32, tn * 16, lane, acc[t]);
    __syncthreads();
  }

  // write out[b, i]
  {
    int r  = tid >> 2;
    int c0 = (tid & 3) * 16;
    float4* dst = (float4*)(out + (size_t)blockIdx.x * (MAT_N * MAT_N) + r * MAT_N + c0);
#pragma unroll
    for (int q = 0; q < 4; ++q) {
      float4 v;
      v.x = Rs[r * LD + c0 + q * 4 + 0];
      v.y = Rs[r * LD + c0 + q * 4 + 1];
      v.z = Rs[r * LD + c0 + q * 4 + 2];
      v.w = Rs[r * LD + c0 + q * 4 + 3];
      dst[q] = v;
    }
  }
}

extern "C" void kernel_launch(void* const* d_in, const int* in_sizes, int n_in,
                              void* d_out, int out_size, void* d_ws, size_t ws_size,
                              hipStream_t stream) {
  (void)in_sizes; (void)n_in; (void)out_size; (void)ws_size;
  const float* x = (const float*)d_in[0];
  const float* w = (const float*)d_in[1];
  float* out = (float*)d_out;
  float* ws  = (float*)d_ws;
  float* wsm  = ws;         // 256 floats
  float* logs = ws + 256;   // B*C*64*64 floats (134 MB, L2-resident)

  sparsemax_kernel<<<1, 32, 0, stream>>>(w, wsm);
  logm_kernel<<<BATCH * CH, 256, 0, stream>>>(x, logs);
  expmix_kernel<<<BATCH * CH, 256, 0, stream>>>(logs, wsm, out);
}